// ZeroShotObjectDetectorWithFeedback_74904229642307
// MI455X (gfx1250) — compile-verified
//
#include <hip/hip_runtime.h>
#include <math.h>

// ---------------------------------------------------------------------------
// Batched NMS for MI455X (gfx1250, wave32).
//  Kernel A: per-image 3600x3600 IoU adjacency bitmask, one wave per 16x16
//            tile. Pairwise area-sum term computed on the matrix pipe via
//            V_WMMA_F32_16X16X4_F32 (rank-2 outer product:
//            D[M][N] = area_i[M]*1 + 1*area_j[N]); the D VGPR layout
//            (lane = column N, VGPR v = row v + 8*(lane/16)) feeds the
//            min/max IoU finish, and one wave32 ballot per VGPR row-pair
//            yields the 16-bit row masks (no LDS atomics). 6.5 MB -> L2.
//  Kernel B: one 256-thread workgroup per (image,class): adj-masked max of
//            con, validity bits via per-wave ballot, stable descending
//            bitonic sort of packed (score,~idx) keys in LDS, sequential
//            greedy suppression over the 113-word valid bitmask
//            (global_prefetch_b8 hides the next candidate's adjacency row),
//            then prefix-sum rank assignment and output writes (suppressed
//            tail is all zeros, matching the reference's zeroed+permuted
//            layout).
// ---------------------------------------------------------------------------

typedef __attribute__((ext_vector_type(2))) float v2f;
typedef __attribute__((ext_vector_type(8))) float v8f;

#define N_BOXES 3600
#define N_CLS   16
#define TILES   225            // 3600 / 16
#define NWORDS  113            // ceil(3600 / 32)
#define ROWB    452            // bitmask row stride in bytes (113 u32 words)
#define SORTN   4096
#define NMS_THRE 0.4f

__device__ __forceinline__ unsigned f2ord(float f) {
  unsigned b = __float_as_uint(f);
  return (b & 0x80000000u) ? ~b : (b | 0x80000000u);
}
__device__ __forceinline__ float ord2f(unsigned o) {
  unsigned b = (o & 0x80000000u) ? (o ^ 0x80000000u) : ~o;
  return __uint_as_float(b);
}

// ---------------------------------------------------------------------------
// Kernel A: IoU adjacency bitmask. grid = (225*225, B), block = 32 (one wave).
// ---------------------------------------------------------------------------
__global__ void iou_adj_kernel(const float* __restrict__ boxes,
                               unsigned char* __restrict__ adj) {
  __shared__ float sx1[32], sy1[32], sx2[32], sy2[32], sarea[32];

  const int lane = threadIdx.x;
  const int b    = blockIdx.y;
  const int tile = blockIdx.x;
  const int ti   = tile % TILES;
  const int tj   = tile / TILES;
  const int i0   = ti * 16;
  const int j0   = tj * 16;

  const int half = lane >> 4;                       // 0: i-side, 1: j-side
  const int boxIdx = half ? (j0 + (lane - 16)) : (i0 + lane);
  const float4 bx = reinterpret_cast<const float4*>(boxes)[(size_t)b * N_BOXES + boxIdx];
  // Replicate reference op order exactly: x1 = cx - w/2 ; x2 = x1 + w
  float x1 = bx.x - bx.z / 2.0f;
  float y1 = bx.y - bx.w / 2.0f;
  float x2 = x1 + bx.z;
  float y2 = y1 + bx.w;
  sx1[lane] = x1; sy1[lane] = y1; sx2[lane] = x2; sy2[lane] = y2;
  sarea[lane] = (x2 - x1) * (y2 - y1);
  __syncthreads();

  // area_i[M] + area_j[N] on the matrix pipe.
  // A (16x4 f32, documented layout): lanes 0-15 hold row M=lane, VGPR0=K0,
  // VGPR1=K1; lanes 16-31 hold K2,K3 (zeros here).
  // A row M = (area_i[M], 1, 0, 0); B col N = (1, area_j[N], 0, 0)
  //   => D[M][N] = area_i[M] + area_j[N], single rounding (matches jnp).
  const bool lo = (lane < 16);
  v2f A, Bm;
  A.x  = lo ? sarea[lane]             : 0.0f;   // K0: area_i[M]
  A.y  = lo ? 1.0f                    : 0.0f;   // K1: ones
  Bm.x = lo ? 1.0f                    : 0.0f;   // K0 row: ones
  Bm.y = lo ? sarea[16 + (lane & 15)] : 0.0f;   // K1 row: area_j[N]
  v8f C = {};
  v8f D = __builtin_amdgcn_wmma_f32_16x16x4_f32(
      /*neg_a=*/false, A, /*neg_b=*/false, Bm,
      /*c_mod=*/(short)0, C, /*reuse_a=*/false, /*reuse_b=*/false);

  // D layout: lane L -> column N = L%16 ; VGPR v -> row v + 8*(L/16).
  // One ballot per v gives: bits 0-15 = row v, bits 16-31 = row v+8.
  // EXEC is all-ones here (no divergence), as ballot and WMMA require.
  const int n = lane & 15;
  const float jx1 = sx1[16 + n], jy1 = sy1[16 + n];
  const float jx2 = sx2[16 + n], jy2 = sy2[16 + n];
  unsigned bal[8];
#pragma unroll
  for (int v = 0; v < 8; ++v) {
    const int r = v + 8 * half;
    const float ltx = fmaxf(sx1[r], jx1);
    const float lty = fmaxf(sy1[r], jy1);
    const float rbx = fminf(sx2[r], jx2);
    const float rby = fminf(sy2[r], jy2);
    const float iw = fmaxf(rbx - ltx, 0.0f);
    const float ih = fmaxf(rby - lty, 0.0f);
    const float inter = iw * ih;
    const float uni = D[v] - inter;
    const bool pred = (inter / uni >= NMS_THRE) && ((i0 + r) != (j0 + n));
    bal[v] = __builtin_amdgcn_ballot_w32(pred);   // wave-uniform row-pair mask
  }

  if (lane < 16) {
    // Select the uniform mask for this output row via a cndmask chain
    // (bal[] values live in SGPRs; avoid a scratch-backed dynamic index).
    const unsigned sel = (unsigned)(lane & 7);
    unsigned m = bal[0];
    m = (sel == 1u) ? bal[1] : m;
    m = (sel == 2u) ? bal[2] : m;
    m = (sel == 3u) ? bal[3] : m;
    m = (sel == 4u) ? bal[4] : m;
    m = (sel == 5u) ? bal[5] : m;
    m = (sel == 6u) ? bal[6] : m;
    m = (sel == 7u) ? bal[7] : m;
    const unsigned short hw =
        (lane < 8) ? (unsigned short)(m & 0xFFFFu) : (unsigned short)(m >> 16);
    const size_t rowBase = ((size_t)b * N_BOXES + (i0 + lane)) * (size_t)ROWB;
    *reinterpret_cast<unsigned short*>(adj + rowBase + (j0 >> 3)) = hw;
  }
}

// ---------------------------------------------------------------------------
// Kernel B: per (image,class) NMS. grid = (C, B), block = 256.
// ---------------------------------------------------------------------------
__global__ void nms_kernel(const float* __restrict__ pro,
                           const float* __restrict__ con,
                           const float* __restrict__ boxes,
                           const float* __restrict__ scales,
                           const float* __restrict__ conf_thres,
                           const unsigned char* __restrict__ adj,
                           float* __restrict__ out, int B) {
  __shared__ float s_con[N_BOXES];                 // 14400 B
  __shared__ unsigned long long s_key[SORTN];      // 32768 B
  __shared__ unsigned s_valid[128];                //   512 B
  __shared__ unsigned s_cnt[256];                  //  1024 B

  const int tid = threadIdx.x;
  const int c = blockIdx.x;
  const int b = blockIdx.y;
  const float conf  = conf_thres[c];
  const float scale = scales[b];

  for (int j = tid; j < N_BOXES; j += 256)
    s_con[j] = con[((size_t)b * N_BOXES + j) * N_CLS + c];
  if (tid < 128) s_valid[tid] = 0u;
  __syncthreads();

  // adj_con = max over adjacent j of con_c[j]; valid0 test; sort keys.
  // Lanes of one wave cover 32 consecutive i -> one ballot per iteration
  // produces a full s_valid word; lane 0 stores it (no atomics). On the
  // ragged last iteration inactive lanes contribute 0 bits and untouched
  // words keep their zero init.
  for (int i = tid; i < N_BOXES; i += 256) {
    const unsigned* row =
        reinterpret_cast<const unsigned*>(adj + ((size_t)b * N_BOXES + i) * (size_t)ROWB);
    float m = 0.0f;
    for (int w = 0; w < NWORDS; ++w) {
      unsigned u = row[w];
      if (w == NWORDS - 1) u &= 0xFFFFu;           // bits beyond N are unwritten
      while (u) {
        const int t = __builtin_ctz(u);
        m = fmaxf(m, s_con[(w << 5) + t]);
        u &= u - 1;
      }
    }
    const float p = pro[((size_t)b * N_BOXES + i) * N_CLS + c];
    const bool v0 = (p >= conf) && (p >= m) && (p >= s_con[i]);
    const unsigned bw = __builtin_amdgcn_ballot_w32(v0);
    if ((tid & 31) == 0) s_valid[i >> 5] = bw;
    // key: score (order-preserving bits) desc, ties -> smaller index first
    s_key[i] = ((unsigned long long)f2ord(p) << 32) |
               (unsigned long long)(0xFFFFFFFFu - (unsigned)i);
  }
  for (int i = N_BOXES + tid; i < SORTN; i += 256) s_key[i] = 0ull;  // pad sorts last
  __syncthreads();

  // Bitonic sort, descending, 4096 keys in LDS.
  for (unsigned k = 2; k <= SORTN; k <<= 1) {
    for (unsigned j = k >> 1; j > 0; j >>= 1) {
      for (unsigned e = tid; e < SORTN; e += 256) {
        const unsigned p = e ^ j;
        if (p > e) {
          const unsigned long long a = s_key[e], q = s_key[p];
          const bool doSwap = ((e & k) == 0) ? (a < q) : (a > q);
          if (doSwap) { s_key[e] = q; s_key[p] = a; }
        }
      }
      __syncthreads();
    }
  }

  // Greedy suppression in sorted order. The candidate's own adjacency bit is
  // zero (no diagonal), so the check bit is invariant under this iteration's
  // writes -> one barrier per iteration is sufficient.
  for (int k = 0; k < N_BOXES; ++k) {
    __syncthreads();
    const unsigned long long kk = s_key[k];
    const unsigned idx = 0xFFFFFFFFu - (unsigned)kk;
    const unsigned wv = s_valid[idx >> 5];
    if ((wv >> (idx & 31)) & 1u) {
      const unsigned* row = reinterpret_cast<const unsigned*>(
          adj + ((size_t)b * N_BOXES + idx) * (size_t)ROWB);
      if (tid < NWORDS) s_valid[tid] &= ~row[tid];
    }
    if (tid == 0 && (k + 1) < N_BOXES) {
      const unsigned nid = 0xFFFFFFFFu - (unsigned)s_key[k + 1];
      __builtin_prefetch(adj + ((size_t)b * N_BOXES + nid) * (size_t)ROWB, 0, 0);
    }
  }
  __syncthreads();

  // Outputs: bws (B,C,N,5) then mask (B,C,N). Suppressed tail = zeros.
  float* outBws  = out + (size_t)(b * N_CLS + c) * N_BOXES * 5;
  float* outMask = out + (size_t)B * N_CLS * N_BOXES * 5 +
                   (size_t)(b * N_CLS + c) * N_BOXES;
  for (int t = tid; t < N_BOXES * 5; t += 256) outBws[t] = 0.0f;
  for (int t = tid; t < N_BOXES; t += 256) outMask[t] = 0.0f;

  // Rank assignment: blocked prefix sum over sorted-order survivor flags.
  const int CH = (N_BOXES + 255) / 256;            // 15
  const int k0 = tid * CH;
  const int k1 = min(N_BOXES, k0 + CH);
  unsigned cnt = 0;
  for (int k = k0; k < k1; ++k) {
    const unsigned idx = 0xFFFFFFFFu - (unsigned)s_key[k];
    cnt += (s_valid[idx >> 5] >> (idx & 31)) & 1u;
  }
  s_cnt[tid] = cnt;
  __syncthreads();
  for (int off = 1; off < 256; off <<= 1) {        // Hillis-Steele inclusive scan
    const unsigned v = (tid >= off) ? s_cnt[tid - off] : 0u;
    __syncthreads();
    s_cnt[tid] += v;
    __syncthreads();
  }
  unsigned r = s_cnt[tid] - cnt;                   // exclusive prefix

  for (int k = k0; k < k1; ++k) {
    const unsigned long long kk = s_key[k];
    const unsigned idx = 0xFFFFFFFFu - (unsigned)kk;
    if ((s_valid[idx >> 5] >> (idx & 31)) & 1u) {
      const float4 bx =
          reinterpret_cast<const float4*>(boxes)[(size_t)b * N_BOXES + idx];
      // reference: b = boxes * scale first, then cx - 0.5*w etc.
      const float cx = bx.x * scale, cy = bx.y * scale;
      const float w  = bx.z * scale, h  = bx.w * scale;
      float* o = outBws + (size_t)r * 5;
      o[0] = cx - 0.5f * w;
      o[1] = cy - 0.5f * h;
      o[2] = cx + 0.5f * w;
      o[3] = cy + 0.5f * h;
      o[4] = ord2f((unsigned)(kk >> 32));
      outMask[r] = 1.0f;
      ++r;
    }
  }
}

// ---------------------------------------------------------------------------
extern "C" void kernel_launch(void* const* d_in, const int* in_sizes, int n_in,
                              void* d_out, int out_size, void* d_ws, size_t ws_size,
                              hipStream_t stream) {
  const float* pro    = (const float*)d_in[0];   // (B,N,C)
  const float* con    = (const float*)d_in[1];   // (B,N,C)
  const float* boxes  = (const float*)d_in[2];   // (B,N,4)
  const float* scales = (const float*)d_in[3];   // (B,)
  const float* conf   = (const float*)d_in[4];   // (C,)
  const int B = in_sizes[3];                     // 4

  // Adjacency bitmask in scratch: B * 3600 * 452 bytes (~6.5 MB, L2-resident).
  dim3 gA(TILES * TILES, B);
  iou_adj_kernel<<<gA, 32, 0, stream>>>(boxes, (unsigned char*)d_ws);

  dim3 gB(N_CLS, B);
  nms_kernel<<<gB, 256, 0, stream>>>(pro, con, boxes, scales, conf,
                                     (const unsigned char*)d_ws,
                                     (float*)d_out, B);
}